// GATv2Layer_67216238182418
// MI455X (gfx1250) — compile-verified
//
#include <hip/hip_runtime.h>
#include <hip/hip_bf16.h>
#include <math.h>

typedef __attribute__((ext_vector_type(2))) float v2f;
typedef __attribute__((ext_vector_type(8))) float v8f;

#define N_NODES 50000
#define N_EDGES 800000
#define IN_DIM  128
#define HEADS   4
#define OUT_DIM 16
#define HD      64
#define NEG_SLOPE 0.2f

// ---------------- projection GEMM: x@[Wl|Wr|Wres] + bias via V_WMMA_F32_16X16X4_F32 ----
__global__ __launch_bounds__(128) void proj_gemm(
    const float* __restrict__ x,
    const float* __restrict__ Wl,  const float* __restrict__ bl,
    const float* __restrict__ Wr,  const float* __restrict__ br,
    const float* __restrict__ Wrs, const float* __restrict__ brs,
    float* __restrict__ xl, float* __restrict__ xr, float* __restrict__ rs)
{
    const int lane = threadIdx.x & 31;
    const int wave = threadIdx.x >> 5;          // 0..3 -> 16-col slice
    const int mt   = blockIdx.x;                // 3125 tiles * 16 rows = 50000
    const int rowA = mt * 16 + (lane & 15);
    const int koff = (lane >> 4) * 2;           // A/B: lanes 16-31 hold K+2,K+3
    const int col  = lane & 15;
    const int c    = wave * 16 + col;           // global output column 0..63

    v8f a0 = {}, a1 = {}, a2 = {};
    const float* xrow = x + (size_t)rowA * IN_DIM;

    #pragma unroll 4
    for (int k0 = 0; k0 < IN_DIM; k0 += 4) {
        const int kw = k0 + koff;
        v2f a = *(const v2f*)(xrow + kw);       // A 16x4 f32 fragment (2 VGPRs/lane)
        v2f b0, b1, b2;                         // B 4x16 f32 fragments
        b0.x = Wl [(size_t)kw * HD + c];  b0.y = Wl [(size_t)(kw + 1) * HD + c];
        b1.x = Wr [(size_t)kw * HD + c];  b1.y = Wr [(size_t)(kw + 1) * HD + c];
        b2.x = Wrs[(size_t)kw * HD + c];  b2.y = Wrs[(size_t)(kw + 1) * HD + c];
        a0 = __builtin_amdgcn_wmma_f32_16x16x4_f32(false, a, false, b0, (short)0, a0, false, false);
        a1 = __builtin_amdgcn_wmma_f32_16x16x4_f32(false, a, false, b1, (short)0, a1, false, false);
        a2 = __builtin_amdgcn_wmma_f32_16x16x4_f32(false, a, false, b2, (short)0, a2, false, false);
    }

    const float vbl = bl[c], vbr = br[c], vbs = brs[c];
    #pragma unroll
    for (int i = 0; i < 8; ++i) {
        const int row = mt * 16 + i + 8 * (lane >> 4);   // C/D layout, ISA 7.12.2
        const size_t o = (size_t)row * HD + c;
        xl[o] = a0[i] + vbl;
        xr[o] = a1[i] + vbr;
        rs[o] = a2[i] + vbs;
    }
}

// ---------------- helpers ----------------
__device__ inline void atomicMaxFloat(float* addr, float val) {
    // valid when location initialized to -inf
    if (val >= 0.f) atomicMax((int*)addr, __float_as_int(val));
    else            atomicMin((unsigned int*)addr, (unsigned int)__float_as_int(val));
}

// zero out-accumulator (lives in d_out), init m=-inf, denom=0
__global__ void init_kernel(float* __restrict__ acc, float* __restrict__ m,
                            float* __restrict__ denom)
{
    const int i = blockIdx.x * blockDim.x + threadIdx.x;
    if (i < N_NODES * HD) acc[i] = 0.f;
    if (i < N_NODES * HEADS) { m[i] = -INFINITY; denom[i] = 0.f; }
}

// ---------------- per-(edge,head) score + segment max ----------------
__global__ __launch_bounds__(256) void edge_score(
    const float* __restrict__ xl, const float* __restrict__ xr,
    const long long* __restrict__ ei, const float* __restrict__ ea,
    const float* __restrict__ We, const float* __restrict__ att,
    float* __restrict__ score, float* __restrict__ m)
{
    const int t = blockIdx.x * blockDim.x + threadIdx.x;
    if (t >= N_EDGES * HEADS) return;
    const int e = t >> 2, h = t & 3;
    const long long s = ei[e], d = ei[N_EDGES + e];
    const float eatt = ea[e];
    const float4* pj = (const float4*)(xl + (size_t)s * HD + h * OUT_DIM);
    const float4* pr = (const float4*)(xr + (size_t)d * HD + h * OUT_DIM);
    const float4* pw = (const float4*)(We + h * OUT_DIM);
    const float4* pa = (const float4*)(att + h * OUT_DIM);
    float sc = 0.f;
    #pragma unroll
    for (int q = 0; q < 4; ++q) {
        const float4 fj = pj[q], fr = pr[q], fw = pw[q], fa = pa[q];
        float f;
        f = fj.x + fr.x + eatt * fw.x; f = f > 0.f ? f : NEG_SLOPE * f; sc += fa.x * f;
        f = fj.y + fr.y + eatt * fw.y; f = f > 0.f ? f : NEG_SLOPE * f; sc += fa.y * f;
        f = fj.z + fr.z + eatt * fw.z; f = f > 0.f ? f : NEG_SLOPE * f; sc += fa.z * f;
        f = fj.w + fr.w + eatt * fw.w; f = f > 0.f ? f : NEG_SLOPE * f; sc += fa.w * f;
    }
    score[t] = sc;
    atomicMaxFloat(&m[(size_t)d * HEADS + h], sc);
}

// ---------------- exp + segment sum ----------------
__global__ __launch_bounds__(256) void edge_exp(
    const long long* __restrict__ ei, const float* __restrict__ score,
    const float* __restrict__ m, float* __restrict__ exs,
    float* __restrict__ denom)
{
    const int t = blockIdx.x * blockDim.x + threadIdx.x;
    if (t >= N_EDGES * HEADS) return;
    const int e = t >> 2, h = t & 3;
    const long long d = ei[N_EDGES + e];
    const float ex = expf(score[t] - m[(size_t)d * HEADS + h]);
    exs[t] = ex;
    atomicAdd(&denom[(size_t)d * HEADS + h], ex);
}

// ---------------- alpha + scatter-add aggregation ----------------
__global__ __launch_bounds__(256) void edge_aggr(
    const long long* __restrict__ ei, const float* __restrict__ exs,
    const float* __restrict__ denom, const float* __restrict__ xl,
    float* __restrict__ alpha_out, float* __restrict__ acc)
{
    const int t = blockIdx.x * blockDim.x + threadIdx.x;
    if (t >= N_EDGES * HEADS) return;
    const int e = t >> 2, h = t & 3;
    const long long s = ei[e], d = ei[N_EDGES + e];
    const float alpha = exs[t] / (denom[(size_t)d * HEADS + h] + 1e-16f);
    alpha_out[t] = alpha;
    const float4* pj = (const float4*)(xl + (size_t)s * HD + h * OUT_DIM);
    float* ap = acc + (size_t)d * HD + h * OUT_DIM;
    #pragma unroll
    for (int q = 0; q < 4; ++q) {
        const float4 fj = pj[q];
        atomicAdd(ap + 4 * q + 0, alpha * fj.x);
        atomicAdd(ap + 4 * q + 1, alpha * fj.y);
        atomicAdd(ap + 4 * q + 2, alpha * fj.z);
        atomicAdd(ap + 4 * q + 3, alpha * fj.w);
    }
}

// ---------------- bias + residual + ELU (in place on d_out region) ----------------
__global__ void finalize_kernel(float* __restrict__ acc,
                                const float* __restrict__ bias_out,
                                const float* __restrict__ rs)
{
    const int i = blockIdx.x * blockDim.x + threadIdx.x;
    if (i >= N_NODES * HD) return;
    float v = acc[i] + bias_out[i & (HD - 1)] + rs[i];
    acc[i] = v > 0.f ? v : (expf(v) - 1.f);
}

// ---------------- passthrough edge_index into d_out ----------------
__global__ void copy_ei(const long long* __restrict__ ei, long long* __restrict__ dst)
{
    const int i = blockIdx.x * blockDim.x + threadIdx.x;
    if (i < 2 * N_EDGES) dst[i] = ei[i];
}

extern "C" void kernel_launch(void* const* d_in, const int* in_sizes, int n_in,
                              void* d_out, int out_size, void* d_ws, size_t ws_size,
                              hipStream_t stream) {
    (void)in_sizes; (void)n_in; (void)out_size; (void)ws_size;
    const float*     x   = (const float*)d_in[0];
    const long long* ei  = (const long long*)d_in[1];
    const float*     ea  = (const float*)d_in[2];
    const float*     Wl  = (const float*)d_in[3];
    const float*     bl  = (const float*)d_in[4];
    const float*     Wr  = (const float*)d_in[5];
    const float*     br  = (const float*)d_in[6];
    const float*     We  = (const float*)d_in[7];
    const float*     att = (const float*)d_in[8];
    const float*     bo  = (const float*)d_in[9];
    const float*     Wrs = (const float*)d_in[10];
    const float*     brs = (const float*)d_in[11];

    // workspace layout (floats)
    float* ws    = (float*)d_ws;
    float* xl    = ws;                               // N*HD
    float* xr    = xl + (size_t)N_NODES * HD;        // N*HD
    float* rs    = xr + (size_t)N_NODES * HD;        // N*HD
    float* score = rs + (size_t)N_NODES * HD;        // E*H
    float* exs   = score + (size_t)N_EDGES * HEADS;  // E*H
    float* m     = exs + (size_t)N_EDGES * HEADS;    // N*H
    float* denom = m + (size_t)N_NODES * HEADS;      // N*H

    // output layout: out [N*HD] f32 | edge_index [2E] i64 | alpha [E*H] f32
    float*     out_acc   = (float*)d_out;
    long long* out_ei    = (long long*)((char*)d_out + (size_t)N_NODES * HD * sizeof(float));
    float*     out_alpha = (float*)((char*)out_ei + (size_t)2 * N_EDGES * sizeof(long long));

    const int EH = N_EDGES * HEADS;

    proj_gemm<<<N_NODES / 16, 128, 0, stream>>>(x, Wl, bl, Wr, br, Wrs, brs, xl, xr, rs);
    init_kernel<<<(N_NODES * HD + 255) / 256, 256, 0, stream>>>(out_acc, m, denom);
    edge_score<<<(EH + 255) / 256, 256, 0, stream>>>(xl, xr, ei, ea, We, att, score, m);
    edge_exp<<<(EH + 255) / 256, 256, 0, stream>>>(ei, score, m, exs, denom);
    edge_aggr<<<(EH + 255) / 256, 256, 0, stream>>>(ei, exs, denom, xl, out_alpha, out_acc);
    finalize_kernel<<<(N_NODES * HD + 255) / 256, 256, 0, stream>>>(out_acc, bo, rs);
    copy_ei<<<(2 * N_EDGES + 255) / 256, 256, 0, stream>>>(ei, out_ei);
}